// DN1_83434034692098
// MI455X (gfx1250) — compile-verified
//
#include <hip/hip_runtime.h>
#include <math.h>

typedef __attribute__((ext_vector_type(2))) float v2f;
typedef __attribute__((ext_vector_type(8))) float v8f;

#define THRESH 0.5f

// ---------------------------------------------------------------------------
// Kernel 0: build inpt = concat(x, y_response, z); zero s_y/s_z accumulators.
// ---------------------------------------------------------------------------
__global__ __launch_bounds__(256)
void k_prep(const float* __restrict__ x, const float* __restrict__ y,
            const float* __restrict__ z, int x_size, int ny, int zs,
            float* __restrict__ inpt, float* __restrict__ s_all, int s_total) {
  int i = blockIdx.x * blockDim.x + threadIdx.x;
  int cols = x_size + ny + zs;
  if (i < cols) {
    float v;
    if (i < x_size)            v = x[i];
    else if (i < x_size + ny)  v = y[i - x_size];
    else                       v = z[i - x_size - ny];
    inpt[i] = v;
  }
  if (i < s_total) s_all[i] = 0.f;
}

// ---------------------------------------------------------------------------
// Kernel 1: response = neurons @ inpt via V_WMMA_F32_16X16X4_F32.
// One block per 16-row strip; 8 waves split the K dimension (cols/8 each).
// A tile layout (16x4 f32): lane m (0..15) holds row M=m; VGPR j, lane-half h
//   -> K = j + 2h  => one b64 load per lane per K-chunk of 4.
// B (4x16) is inpt replicated across all N columns with the same K = j + 2h
// mapping, so every D column equals the row dot product.
// K loop unrolled x4 (16 K per iter) with two alternating accumulators to
// break the WMMA C-dependency chain and let loads run ahead of the waits.
// ---------------------------------------------------------------------------
__global__ __launch_bounds__(256)
void k_response(const float* __restrict__ neurons,
                const float* __restrict__ inpt,
                const float* __restrict__ ages,
                int cols, int ny,
                float* __restrict__ coef, float* __restrict__ act,
                float* __restrict__ out_z, float* __restrict__ out_ages) {
  __shared__ float lds[128];                 // 8 waves x 16 row-partials
  const int tid  = threadIdx.x;
  const int wave = tid >> 5;
  const int lane = tid & 31;
  const int h    = lane >> 4;                // lane-half
  const int m    = lane & 15;                // row within strip
  const int row0 = blockIdx.x * 16;
  const int kseg = cols >> 3;                // K per wave (1152), mult of 16
  const int kbeg = wave * kseg;

  const float* aptr = neurons + (size_t)(row0 + m) * (size_t)cols + 2 * h;
  const float* bptr = inpt + 2 * h;

  v8f c0 = {0.f, 0.f, 0.f, 0.f, 0.f, 0.f, 0.f, 0.f};
  v8f c1 = {0.f, 0.f, 0.f, 0.f, 0.f, 0.f, 0.f, 0.f};
  for (int k = kbeg; k < kbeg + kseg; k += 16) {
    v2f a0 = *(const v2f*)(aptr + k +  0);
    v2f b0 = *(const v2f*)(bptr + k +  0);
    v2f a1 = *(const v2f*)(aptr + k +  4);
    v2f b1 = *(const v2f*)(bptr + k +  4);
    v2f a2 = *(const v2f*)(aptr + k +  8);
    v2f b2 = *(const v2f*)(bptr + k +  8);
    v2f a3 = *(const v2f*)(aptr + k + 12);
    v2f b3 = *(const v2f*)(bptr + k + 12);
    c0 = __builtin_amdgcn_wmma_f32_16x16x4_f32(false, a0, false, b0,
                                               (short)0, c0, false, false);
    c1 = __builtin_amdgcn_wmma_f32_16x16x4_f32(false, a1, false, b1,
                                               (short)0, c1, false, false);
    c0 = __builtin_amdgcn_wmma_f32_16x16x4_f32(false, a2, false, b2,
                                               (short)0, c0, false, false);
    c1 = __builtin_amdgcn_wmma_f32_16x16x4_f32(false, a3, false, b3,
                                               (short)0, c1, false, false);
  }
  v8f c = c0 + c1;                           // element-wise == matrix add

  // D layout: VGPR v, lanes 0-15 -> M=v ; lanes 16-31 -> M=v+8 (N = any).
  if (m == 0) {
    float* dst = lds + wave * 16 + h * 8;
#pragma unroll
    for (int v = 0; v < 8; ++v) dst[v] = c[v];
  }
  __syncthreads();

  if (tid < 16) {
    float dot = 0.f;
#pragma unroll
    for (int w = 0; w < 8; ++w) dot += lds[w * 16 + tid];
    int r = row0 + tid;
    float resp = dot - THRESH;
    bool on = resp > 0.f;
    float age = ages[r];
    coef[r] = on ? (age - 1.f) / age : 0.f;
    act[r]  = on ? 1.f : 0.f;
    out_ages[r] = age + (on ? 1.f : 0.f);
    if (r >= ny) out_z[r - ny] = on ? resp : 0.f;   // relu(response)
  }
}

// ---------------------------------------------------------------------------
// Kernel 2: s_g[c] = sum_r coef[r] * neurons[r][c] over each row group.
// Row tiles of 64 never straddle the y/z boundary (4096 % 64 == 0), so the
// destination (s_y vs s_z) is uniform per block. float4 coalesced loads,
// per-block register accumulation, atomicAdd partials.
// ---------------------------------------------------------------------------
__global__ __launch_bounds__(256)
void k_colsum(const float* __restrict__ neurons,
              const float* __restrict__ coef,
              int cols, int ny, int rows_per_blk,
              float* __restrict__ s_y, float* __restrict__ s_z) {
  const int col = blockIdx.x * 1024 + threadIdx.x * 4;
  const int r0  = blockIdx.y * rows_per_blk;
  float4 acc = make_float4(0.f, 0.f, 0.f, 0.f);
  for (int rr = 0; rr < rows_per_blk; ++rr) {
    int r = r0 + rr;
    float cf = coef[r];
    if (cf != 0.f) {
      float4 n = *(const float4*)(neurons + (size_t)r * (size_t)cols + col);
      acc.x += cf * n.x; acc.y += cf * n.y;
      acc.z += cf * n.z; acc.w += cf * n.w;
    }
  }
  float* s = (r0 < ny) ? s_y : s_z;
  atomicAdd(s + col + 0, acc.x);
  atomicAdd(s + col + 1, acc.y);
  atomicAdd(s + col + 2, acc.z);
  atomicAdd(s + col + 3, acc.w);
}

// ---------------------------------------------------------------------------
// Kernel 3: per-row rewrite. Active rows: new = s_g + inpt/age, L2-normalize;
// inactive rows: copy original. One block (8 waves) per row, float4 I/O.
// ---------------------------------------------------------------------------
__global__ __launch_bounds__(256)
void k_update(const float* __restrict__ neurons,
              const float* __restrict__ inpt,
              const float* __restrict__ ages,
              const float* __restrict__ act,
              const float* __restrict__ s_y,
              const float* __restrict__ s_z,
              int cols, int ny,
              float* __restrict__ out_neurons) {
  const int r   = blockIdx.x;
  const int tid = threadIdx.x;
  const int nIter = cols >> 10;                 // 1024 cols per iteration (=9)
  const size_t rowOff = (size_t)r * (size_t)cols;
  __shared__ float red[8];

  if (act[r] > 0.5f) {
    const float* s = (r < ny) ? s_y : s_z;
    const float inv_age = 1.f / ages[r];
    float4 vals[12];
    float sq = 0.f;
    for (int j = 0; j < nIter; ++j) {
      int c0 = j * 1024 + tid * 4;
      float4 sv = *(const float4*)(s + c0);
      float4 iv = *(const float4*)(inpt + c0);
      float4 v;
      v.x = sv.x + iv.x * inv_age;
      v.y = sv.y + iv.y * inv_age;
      v.z = sv.z + iv.z * inv_age;
      v.w = sv.w + iv.w * inv_age;
      vals[j] = v;
      sq += v.x * v.x + v.y * v.y + v.z * v.z + v.w * v.w;
    }
#pragma unroll
    for (int off = 16; off > 0; off >>= 1) sq += __shfl_xor(sq, off, 32);
    if ((tid & 31) == 0) red[tid >> 5] = sq;
    __syncthreads();
    float tot = 0.f;
#pragma unroll
    for (int w = 0; w < 8; ++w) tot += red[w];
    float scale = 1.f / (sqrtf(tot) + 1e-10f);
    for (int j = 0; j < nIter; ++j) {
      int c0 = j * 1024 + tid * 4;
      float4 v = vals[j];
      v.x *= scale; v.y *= scale; v.z *= scale; v.w *= scale;
      *(float4*)(out_neurons + rowOff + c0) = v;
    }
  } else {
    for (int j = 0; j < nIter; ++j) {
      int c0 = j * 1024 + tid * 4;
      *(float4*)(out_neurons + rowOff + c0) =
          *(const float4*)(neurons + rowOff + c0);
    }
  }
}

// ---------------------------------------------------------------------------
extern "C" void kernel_launch(void* const* d_in, const int* in_sizes, int n_in,
                              void* d_out, int out_size, void* d_ws, size_t ws_size,
                              hipStream_t stream) {
  const float* x       = (const float*)d_in[0];
  const float* yresp   = (const float*)d_in[1];
  const float* z       = (const float*)d_in[2];
  const float* neurons = (const float*)d_in[3];
  const float* ages    = (const float*)d_in[4];

  const int x_size = in_sizes[0];          // 4096
  const int ny     = in_sizes[1];          // 4096
  const int zs     = in_sizes[2];          // 1024
  const int rows   = ny + zs;              // 5120
  const int cols   = x_size + ny + zs;     // 9216

  float* out         = (float*)d_out;
  float* out_z       = out;                                    // [zs]
  float* out_neurons = out + zs;                               // [rows*cols]
  float* out_ages    = out + zs + (size_t)rows * (size_t)cols; // [rows]

  float* ws   = (float*)d_ws;
  float* inpt = ws;                 // [cols]
  float* coef = inpt + cols;        // [rows]
  float* act  = coef + rows;        // [rows]
  float* s_y  = act + rows;         // [cols]
  float* s_z  = s_y + cols;         // [cols]  (contiguous with s_y)

  const int s_total = 2 * cols;
  const int prepN   = (s_total > cols) ? s_total : cols;
  k_prep<<<(prepN + 255) / 256, 256, 0, stream>>>(
      x, yresp, z, x_size, ny, zs, inpt, s_y, s_total);

  k_response<<<rows / 16, 256, 0, stream>>>(
      neurons, inpt, ages, cols, ny, coef, act, out_z, out_ages);

  dim3 g2(cols / 1024, rows / 64);
  k_colsum<<<g2, 256, 0, stream>>>(neurons, coef, cols, ny, 64, s_y, s_z);

  k_update<<<rows, 256, 0, stream>>>(
      neurons, inpt, ages, act, s_y, s_z, cols, ny, out_neurons);
}